// HashEncoder_88837103551050
// MI455X (gfx1250) — compile-verified
//
#include <hip/hip_runtime.h>

// ---------------------------------------------------------------------------
// InstantNGP hash-grid encoder for MI455X (gfx1250, wave32).
// D=3, L=16, C=2, base=16, per-level-scale=2, T=2^19, B=2^21.
// Levels 0..2 dense (16^3, 32^3, 64^3), levels 3..15 hashed into 2^19 entries.
// Strategy: table (~57MB) is L2-resident (192MB L2); stream inputs/outputs
// with non-temporal hints so 2.1GB of 8B gathers never leave L2. Fully
// unrolled level loop -> up to 128 independent global_load_b64 per thread.
// ---------------------------------------------------------------------------

typedef float v4f __attribute__((ext_vector_type(4)));

namespace {

constexpr int kL = 16;
constexpr unsigned kTMask = (1u << 19) - 1u;       // T - 1
constexpr unsigned kP1 = 2654435761u;              // prime for y
constexpr unsigned kP2 = 805459861u;               // prime for z

// res[l] = 16 << l  (ceil(16 * 2^l) exactly)
__device__ constexpr int kRes[kL] = {
    16, 32, 64, 128, 256, 512, 1024, 2048,
    4096, 8192, 16384, 32768, 65536, 131072, 262144, 524288};

// offsets in float2 (entry) units; levels 0..2 dense, 3..15 size T=524288
__device__ constexpr int kOff[kL] = {
    0,        4096,     36864,    299008,
    823296,   1347584,  1871872,  2396160,
    2920448,  3444736,  3969024,  4493312,
    5017600,  5541888,  6066176,  6590464};

__device__ __forceinline__ int imin(int a, int b) { return a < b ? a : b; }

} // namespace

extern "C" __global__ __launch_bounds__(256)
void hash_encode_gfx1250(const float* __restrict__ in,   // [B,3]
                         const float* __restrict__ emb,  // [TOTAL,2]
                         float* __restrict__ out,        // [B,32]
                         int n)
{
    const int b = blockIdx.x * 256 + threadIdx.x;
    if (b >= n) return;

    // Streamed, single-use coordinate read: NT hint so it does not displace
    // the L2-resident embedding table.
    const size_t ib = 3u * (size_t)b;
    const float x = __builtin_nontemporal_load(in + ib + 0);
    const float y = __builtin_nontemporal_load(in + ib + 1);
    const float z = __builtin_nontemporal_load(in + ib + 2);

    float acc[2 * kL];

#pragma unroll
    for (int l = 0; l < kL; ++l) {
        const int   res   = kRes[l];
        const float scale = (float)(res - 1);

        const float px = x * scale, py = y * scale, pz = z * scale;
        const float fx = floorf(px), fy = floorf(py), fz = floorf(pz);
        const float tx = px - fx,   ty = py - fy,   tz = pz - fz;

        const int rm1 = res - 1;
        const int ix = (int)fx, iy = (int)fy, iz = (int)fz;
        const int x0 = imin(ix,     rm1), x1 = imin(ix + 1, rm1);
        const int y0 = imin(iy,     rm1), y1 = imin(iy + 1, rm1);
        const int z0 = imin(iz,     rm1), z1 = imin(iz + 1, rm1);

        unsigned i000, i100, i010, i110, i001, i101, i011, i111;
        if (l < 3) {
            // Dense grid: idx = x + y*res + z*res^2 (fits int32 for res<=64)
            const unsigned sy = (unsigned)res;
            const unsigned sz = (unsigned)(res * res);
            const unsigned ux0 = (unsigned)x0,      ux1 = (unsigned)x1;
            const unsigned uy0 = (unsigned)y0 * sy, uy1 = (unsigned)y1 * sy;
            const unsigned uz0 = (unsigned)z0 * sz, uz1 = (unsigned)z1 * sz;
            i000 = ux0 + uy0 + uz0;  i100 = ux1 + uy0 + uz0;
            i010 = ux0 + uy1 + uz0;  i110 = ux1 + uy1 + uz0;
            i001 = ux0 + uy0 + uz1;  i101 = ux1 + uy0 + uz1;
            i011 = ux0 + uy1 + uz1;  i111 = ux1 + uy1 + uz1;
        } else {
            // Spatial hash: (x*1 ^ y*P1 ^ z*P2) & (T-1), uint32 wraparound
            const unsigned hx0 = (unsigned)x0,       hx1 = (unsigned)x1;
            const unsigned hy0 = (unsigned)y0 * kP1, hy1 = (unsigned)y1 * kP1;
            const unsigned hz0 = (unsigned)z0 * kP2, hz1 = (unsigned)z1 * kP2;
            i000 = (hx0 ^ hy0 ^ hz0) & kTMask;  i100 = (hx1 ^ hy0 ^ hz0) & kTMask;
            i010 = (hx0 ^ hy1 ^ hz0) & kTMask;  i110 = (hx1 ^ hy1 ^ hz0) & kTMask;
            i001 = (hx0 ^ hy0 ^ hz1) & kTMask;  i101 = (hx1 ^ hy0 ^ hz1) & kTMask;
            i011 = (hx0 ^ hy1 ^ hz1) & kTMask;  i111 = (hx1 ^ hy1 ^ hz1) & kTMask;
        }

        // 8 independent 8-byte gathers (global_load_b64) — L2 hits.
        const float2* __restrict__ e =
            reinterpret_cast<const float2*>(emb) + kOff[l];
        const float2 v000 = e[i000], v100 = e[i100];
        const float2 v010 = e[i010], v110 = e[i110];
        const float2 v001 = e[i001], v101 = e[i101];
        const float2 v011 = e[i011], v111 = e[i111];

        // Trilinear weights exactly as the reference forms them:
        // w(corner) = prod_d (corner_d ? frac_d : 1-frac_d)
        const float sx = 1.0f - tx, sy = 1.0f - ty, sz = 1.0f - tz;
        float a0 = 0.0f, a1 = 0.0f, w;
        w = sx * sy * sz; a0 = fmaf(w, v000.x, a0); a1 = fmaf(w, v000.y, a1);
        w = tx * sy * sz; a0 = fmaf(w, v100.x, a0); a1 = fmaf(w, v100.y, a1);
        w = sx * ty * sz; a0 = fmaf(w, v010.x, a0); a1 = fmaf(w, v010.y, a1);
        w = tx * ty * sz; a0 = fmaf(w, v110.x, a0); a1 = fmaf(w, v110.y, a1);
        w = sx * sy * tz; a0 = fmaf(w, v001.x, a0); a1 = fmaf(w, v001.y, a1);
        w = tx * sy * tz; a0 = fmaf(w, v101.x, a0); a1 = fmaf(w, v101.y, a1);
        w = sx * ty * tz; a0 = fmaf(w, v011.x, a0); a1 = fmaf(w, v011.y, a1);
        w = tx * ty * tz; a0 = fmaf(w, v111.x, a0); a1 = fmaf(w, v111.y, a1);

        acc[2 * l + 0] = a0;
        acc[2 * l + 1] = a1;
    }

    // 128B per point, 16B-aligned: 8x global_store_b128 with NT hint so the
    // 256MB output stream does not evict the embedding table from L2.
    // Use a native ext_vector type: __builtin_nontemporal_store rejects the
    // HIP_vector_type wrapper class.
    v4f* __restrict__ o = reinterpret_cast<v4f*>(out + (size_t)b * (2 * kL));
    const v4f* av = reinterpret_cast<const v4f*>(acc);
#pragma unroll
    for (int k = 0; k < 8; ++k) {
        __builtin_nontemporal_store(av[k], o + k);
    }
}

extern "C" void kernel_launch(void* const* d_in, const int* in_sizes, int n_in,
                              void* d_out, int out_size, void* d_ws, size_t ws_size,
                              hipStream_t stream) {
    (void)n_in; (void)out_size; (void)d_ws; (void)ws_size;

    const float* inputs     = (const float*)d_in[0];  // [B,3] f32
    const float* embeddings = (const float*)d_in[1];  // [TOTAL,2] f32
    float*       out        = (float*)d_out;          // [B,32] f32

    const int n      = in_sizes[0] / 3;               // B
    const int blocks = (n + 255) / 256;

    hipLaunchKernelGGL(hash_encode_gfx1250, dim3(blocks), dim3(256), 0, stream,
                       inputs, embeddings, out, n);
}